// CastDisjointToBatchedAttributes_16810501996905
// MI455X (gfx1250) — compile-verified
//
#include <hip/hip_runtime.h>
#include <stdint.h>

#define F     256
#define MAXL  1024
#define WAVES_PER_BLOCK 8

typedef float v4f __attribute__((ext_vector_type(4)));
typedef int   v4i __attribute__((ext_vector_type(4)));

// gfx1250 async global<->LDS builtins (ASYNCcnt path). Guarded so the file
// compiles on toolchains that lack them; fallback is plain b128 copies.
#if defined(__has_builtin)
# if __has_builtin(__builtin_amdgcn_global_load_async_to_lds_b128) && \
     __has_builtin(__builtin_amdgcn_global_store_async_from_lds_b128)
#  define USE_ASYNC_LDS 1
# endif
#endif
#ifndef USE_ASYNC_LDS
# define USE_ASYNC_LDS 0
#endif

// gfx12-style CPol: TH in bits[2:0]; TH_NT == 1 (non-temporal, touch-once)
#define CPOL_NT 1

// pointer-to-addrspace-qualified-int4 types (builtin parameter types)
typedef __attribute__((address_space(1))) v4i* gp4;   // global int4*
typedef __attribute__((address_space(3))) v4i* lp4;   // LDS int4*

// ---------------------------------------------------------------------------
// Main kernel (placed first so the disasm snippet shows it): one wave32 per
// output row.  Row = 256 f32 = 1KB = 2 x b128 per lane.  Valid rows stream
// through LDS via the gfx1250 async data mover; pad rows get NT zero stores.
// ---------------------------------------------------------------------------
__global__ __launch_bounds__(32 * WAVES_PER_BLOCK) void gather_rows_kernel(
    const float* __restrict__ attr,
    const int*   __restrict__ attr_len,
    const int*   __restrict__ starts,
    float*       __restrict__ out)
{
#if USE_ASYNC_LDS
    __shared__ float lds[WAVES_PER_BLOCK * F];   // 1KB per wave
#endif
    const int lane = threadIdx.x & 31;
    const int wave = threadIdx.x >> 5;
    const int row  = blockIdx.x * WAVES_PER_BLOCK + wave;   // [0, B*MAXL)
    const int b    = row >> 10;          // MAXL == 1024
    const int m    = row & (MAXL - 1);
    const int glen = attr_len[b];        // uniform per wave

    float* dst = out + (size_t)row * F;

    if (m < glen) {
        const float* src = attr + (size_t)(starts[b] + m) * F;
#if USE_ASYNC_LDS
        float* lbuf = lds + wave * F;
        // per-lane 16B chunks; lanes 0..31 cover 512B, two chunks = 1KB row
        __builtin_amdgcn_global_load_async_to_lds_b128(
            (gp4)(src + lane * 4),       (lp4)(lbuf + lane * 4),       0, CPOL_NT);
        __builtin_amdgcn_global_load_async_to_lds_b128(
            (gp4)(src + lane * 4 + 128), (lp4)(lbuf + lane * 4 + 128), 0, CPOL_NT);
        asm volatile("s_wait_asynccnt 0x0" ::: "memory");   // LDS data ready
        __builtin_amdgcn_global_store_async_from_lds_b128(
            (gp4)(dst + lane * 4),       (lp4)(lbuf + lane * 4),       0, CPOL_NT);
        __builtin_amdgcn_global_store_async_from_lds_b128(
            (gp4)(dst + lane * 4 + 128), (lp4)(lbuf + lane * 4 + 128), 0, CPOL_NT);
        // implicit wait-idle at s_endpgm covers the outstanding async stores
#else
        v4f a0 = __builtin_nontemporal_load((const v4f*)(src + lane * 4));
        v4f a1 = __builtin_nontemporal_load((const v4f*)(src + lane * 4 + 128));
        __builtin_nontemporal_store(a0, (v4f*)(dst + lane * 4));
        __builtin_nontemporal_store(a1, (v4f*)(dst + lane * 4 + 128));
#endif
    } else {
        v4f z = {0.0f, 0.0f, 0.0f, 0.0f};
        __builtin_nontemporal_store(z, (v4f*)(dst + lane * 4));
        __builtin_nontemporal_store(z, (v4f*)(dst + lane * 4 + 128));
    }
}

// ---------------------------------------------------------------------------
// Kernel 2: exclusive prefix sum of attr_len -> starts (B <= 256)
// ---------------------------------------------------------------------------
__global__ __launch_bounds__(256) void scan_starts_kernel(
    const int* __restrict__ len, int* __restrict__ starts, int B)
{
    __shared__ int s[256];
    const int t = threadIdx.x;
    s[t] = (t < B) ? len[t] : 0;
    __syncthreads();
    // Hillis-Steele inclusive scan (8 steps)
    for (int off = 1; off < 256; off <<= 1) {
        int v = 0;
        if (t >= off) v = s[t - off];
        __syncthreads();
        if (t >= off) s[t] += v;
        __syncthreads();
    }
    if (t < B) starts[t] = s[t] - len[t];   // exclusive
}

// ---------------------------------------------------------------------------
extern "C" void kernel_launch(void* const* d_in, const int* in_sizes, int n_in,
                              void* d_out, int out_size, void* d_ws, size_t ws_size,
                              hipStream_t stream)
{
    (void)n_in; (void)out_size; (void)ws_size;
    const float* attr  = (const float*)d_in[0];
    // d_in[1] (graph_id_attr) is redundant: nodes of each graph are contiguous.
    const int*   alen  = (const int*)d_in[2];
    const int    B     = in_sizes[2];            // 256

    int* starts = (int*)d_ws;                    // B ints of scratch

    scan_starts_kernel<<<1, 256, 0, stream>>>(alen, starts, B);

    const int rows   = B * MAXL;                 // 262144 output rows
    const int blocks = rows / WAVES_PER_BLOCK;   // 32768
    gather_rows_kernel<<<blocks, 32 * WAVES_PER_BLOCK, 0, stream>>>(
        attr, alen, starts, (float*)d_out);
}